// CTCDecoder_9182640079288
// MI455X (gfx1250) — compile-verified
//
#include <hip/hip_runtime.h>
#include <math.h>
#include <stdint.h>

// Problem constants (fixed by setup_inputs)
#define NN 16
#define TT 512
#define DD 512
#define VV 4096
#define LL 128
#define SS 257            // 2*L+1
#define MM (NN*TT)        // 8192 GEMM rows

typedef __attribute__((ext_vector_type(2))) float v2f;
typedef __attribute__((ext_vector_type(8))) float v8f;

#define NEGV (-1e30f)

__device__ __forceinline__ float lae(float a, float b) {
    float mx = fmaxf(a, b), mn = fminf(a, b);
    return mx + log1pf(__expf(mn - mx));
}

// --- address-space helpers (generic LDS addr[31:0] == LDS byte offset) -----
// async-load builtin signature: (AS1 v4i* src, AS3 v4i* dst, imm offset, imm cpol)
typedef int v4i_vs __attribute__((vector_size(4 * sizeof(int))));
typedef __attribute__((address_space(1))) v4i_vs* gas_v4i;
typedef __attribute__((address_space(3))) v4i_vs* lds_v4i;
__device__ __forceinline__ gas_v4i as_g(const void* p) {
    return (gas_v4i)(unsigned long long)p;
}
__device__ __forceinline__ lds_v4i as_l(void* p) {
    return (lds_v4i)(unsigned int)(unsigned long long)p;
}

#if __has_builtin(__builtin_amdgcn_global_load_async_to_lds_b128)
#define USE_ASYNC_LDS 1
#if __has_builtin(__builtin_amdgcn_s_wait_asynccnt)
#define WAIT_ASYNC(n) __builtin_amdgcn_s_wait_asynccnt(n)
#else
#define WAIT_ASYNC(n) asm volatile("s_wait_asynccnt %0" :: "n"(n) : "memory")
#endif
#else
#define WAIT_ASYNC(n)
#endif

// ---------------------------------------------------------------------------
// Kernel 0: zero the sumexp accumulator region of the workspace
// ---------------------------------------------------------------------------
__global__ void zero_ws(float* __restrict__ p, int n) {
    int i = blockIdx.x * blockDim.x + threadIdx.x;
    if (i < n) p[i] = 0.0f;
}

// ---------------------------------------------------------------------------
// Kernel 1: fp32 WMMA GEMM (logits = enc @ W + b) with fused exp-row-sum.
// Never writes logits; only atomically accumulates sum(exp(logit)) per row.
// Block: 256 threads = 8 waves; tile BM=128 x BN=64; each wave -> 32x32.
// Double-buffered LDS, async global->LDS staging when available.
// ---------------------------------------------------------------------------
#define BM 128
#define BN 64
#define BK 16
#define APAD 20   // A LDS row stride (floats): 80B rows -> b128-aligned slots
#define BPAD 68   // B LDS row stride (floats): 272B rows -> b128-aligned slots

__device__ __forceinline__ void stage_tiles(
        const float* __restrict__ A, const float* __restrict__ W,
        int rowBase, int colBase, int kt, int tid,
        float (*Asb)[APAD], float (*Bsb)[BPAD])
{
#if defined(USE_ASYNC_LDS)
    // 3 async b128 instructions per wave per stage (2 for A, 1 for B)
#pragma unroll
    for (int rep = 0; rep < 2; ++rep) {
        int idx = tid + rep * 256;              // 0..511
        int m   = idx >> 2;
        int kv  = (idx & 3) << 2;
        __builtin_amdgcn_global_load_async_to_lds_b128(
            as_g(&A[(size_t)(rowBase + m) * DD + kt + kv]),
            as_l(&Asb[m][kv]), 0, 0);
    }
    {
        int k  = tid >> 4;
        int cv = (tid & 15) << 2;
        __builtin_amdgcn_global_load_async_to_lds_b128(
            as_g(&W[(size_t)(kt + k) * VV + colBase + cv]),
            as_l(&Bsb[k][cv]), 0, 0);
    }
#else
#pragma unroll
    for (int rep = 0; rep < 2; ++rep) {
        int idx = tid + rep * 256;
        int m   = idx >> 2;
        int kv  = (idx & 3) << 2;
        const float4 v = *(const float4*)&A[(size_t)(rowBase + m) * DD + kt + kv];
        *(float4*)&Asb[m][kv] = v;
    }
    {
        int k  = tid >> 4;
        int cv = (tid & 15) << 2;
        const float4 v = *(const float4*)&W[(size_t)(kt + k) * VV + colBase + cv];
        *(float4*)&Bsb[k][cv] = v;
    }
#endif
}

__global__ __launch_bounds__(256) void gemm_sumexp(
        const float* __restrict__ A,   // (8192, 512) row-major
        const float* __restrict__ W,   // (512, 4096) row-major
        const float* __restrict__ b,   // (4096)
        float* __restrict__ sumexp)    // (8192) pre-zeroed
{
    __shared__ float As[2][BM][APAD];
    __shared__ float Bs[2][BK][BPAD];

    const int tid   = threadIdx.x;
    const int lane  = tid & 31;
    const int wave  = tid >> 5;
    const int waveM = wave & 3;   // 0..3  -> 32-row group
    const int waveN = wave >> 2;  // 0..1  -> 32-col group
    const int rowBase = blockIdx.y * BM;
    const int colBase = blockIdx.x * BN;

    v8f acc[2][2];
#pragma unroll
    for (int i = 0; i < 2; ++i)
#pragma unroll
        for (int j = 0; j < 2; ++j)
#pragma unroll
            for (int e = 0; e < 8; ++e) acc[i][j][e] = 0.0f;

    const int mr    = lane & 15;        // 0..15 : M (A) / N (B) index
    const int halfk = (lane >> 4) << 1; // 0 or 2 : K sub-pair per ISA layout

    stage_tiles(A, W, rowBase, colBase, 0, tid, As[0], Bs[0]);

    const int nTiles = DD / BK;         // 32
    for (int it = 0; it < nTiles; ++it) {
        const int cur = it & 1;
        if (it + 1 < nTiles) {
            stage_tiles(A, W, rowBase, colBase, (it + 1) * BK, tid,
                        As[cur ^ 1], Bs[cur ^ 1]);
            WAIT_ASYNC(3);   // next stage may stay in flight; current is done
        } else {
            WAIT_ASYNC(0);
        }
        __syncthreads();     // current-stage tile visible to all waves

#pragma unroll
        for (int k0 = 0; k0 < BK; k0 += 4) {
            v2f afrag[2], bfrag[2];
#pragma unroll
            for (int rt = 0; rt < 2; ++rt) {
                int m = waveM * 32 + rt * 16 + mr;
                afrag[rt][0] = As[cur][m][k0 + halfk];
                afrag[rt][1] = As[cur][m][k0 + halfk + 1];
            }
#pragma unroll
            for (int ct = 0; ct < 2; ++ct) {
                int n = waveN * 32 + ct * 16 + mr;
                bfrag[ct][0] = Bs[cur][k0 + halfk][n];
                bfrag[ct][1] = Bs[cur][k0 + halfk + 1][n];
            }
#pragma unroll
            for (int rt = 0; rt < 2; ++rt)
#pragma unroll
                for (int ct = 0; ct < 2; ++ct)
                    acc[rt][ct] = __builtin_amdgcn_wmma_f32_16x16x4_f32(
                        false, afrag[rt], false, bfrag[ct],
                        (short)0, acc[rt][ct], false, false);
        }
        __syncthreads();     // all reads done before this buffer is restaged
    }

    // Epilogue: bias + exp, reduce across the 16 columns per half-wave,
    // one fp32 atomic per row per wave (32 cols; 2 per 64-col tile).
    float bv[2];
#pragma unroll
    for (int ct = 0; ct < 2; ++ct)
        bv[ct] = b[colBase + waveN * 32 + ct * 16 + mr];

    const int hi = lane >> 4;  // C layout: vgpr i holds M=i (lanes 0-15), M=i+8 (16-31)
#pragma unroll
    for (int rt = 0; rt < 2; ++rt) {
        float rs[8];
#pragma unroll
        for (int i = 0; i < 8; ++i) {
            float e = __expf(acc[rt][0][i] + bv[0]) + __expf(acc[rt][1][i] + bv[1]);
#pragma unroll
            for (int off = 8; off; off >>= 1) e += __shfl_xor(e, off, 16);
            rs[i] = e;
        }
        if (mr == 0) {
            int rowb = rowBase + waveM * 32 + rt * 16 + hi * 8;
#pragma unroll
            for (int i = 0; i < 8; ++i) atomicAdd(&sumexp[rowb + i], rs[i]);
        }
    }
}

// ---------------------------------------------------------------------------
// Kernel 2: gathered logits  g[n,t,s] = enc[n,t,:] . W[:, z[n,s]] + b[z]
// z[n,s] = blank(0) for even s, labels[n, s>>1] for odd s.
// One block handles (n, 16 consecutive t); enc rows staged in LDS so each
// scattered W column is read once per 16 timesteps. Columns are software-
// prefetched (global_prefetch) since they are 4B gathers at 16KB stride.
// ---------------------------------------------------------------------------
#define GT 16
__global__ __launch_bounds__(256) void gather_logits(
        const float* __restrict__ enc, const float* __restrict__ W,
        const float* __restrict__ b, const int* __restrict__ labels,
        float* __restrict__ gbuf)      // (N, T, S)
{
    __shared__ float erow[GT][DD];     // 32 KB
    const int n  = blockIdx.y;
    const int t0 = blockIdx.x * GT;

    const float4* src = (const float4*)&enc[((size_t)n * TT + t0) * DD];
    for (int i = threadIdx.x; i < GT * DD / 4; i += 256) {
        float4 v = src[i];
        int tt = i >> 7;               // /(DD/4)
        int kk = (i & 127) << 2;
        *(float4*)&erow[tt][kk] = v;
    }
    __syncthreads();

    for (int s = threadIdx.x; s < SS; s += 256) {
        const int zv = (s & 1) ? labels[n * LL + (s >> 1)] : 0;
        // prefetch the scattered W column (one line per 16 rows)
#pragma unroll
        for (int pk = 0; pk < DD; pk += 16)
            __builtin_prefetch(&W[(size_t)pk * VV + zv], 0, 3);
        float accv[GT];
#pragma unroll
        for (int tt = 0; tt < GT; ++tt) accv[tt] = 0.0f;
        for (int k = 0; k < DD; ++k) {
            float wv = W[(size_t)k * VV + zv];
#pragma unroll
            for (int tt = 0; tt < GT; ++tt) accv[tt] = fmaf(erow[tt][k], wv, accv[tt]);
        }
        const float bb = b[zv];
#pragma unroll
        for (int tt = 0; tt < GT; ++tt)
            gbuf[((size_t)n * TT + t0 + tt) * SS + s] = accv[tt] + bb;
    }
}

// ---------------------------------------------------------------------------
// Kernel 3: CTC forward (alpha) recursion, one block per sample, alpha in LDS.
// log-normalizers precomputed into LDS to keep the sequential loop short.
// ---------------------------------------------------------------------------
__global__ __launch_bounds__(288) void ctc_forward(
        const float* __restrict__ gbuf, const float* __restrict__ sumexp,
        const int* __restrict__ labels, const int* __restrict__ in_lens,
        const int* __restrict__ lab_lens, float* __restrict__ nll)
{
    __shared__ float alpha[2][SS];
    __shared__ float lse[TT];
    const int n = blockIdx.x;
    const int s = threadIdx.x;
    const int len = in_lens[n];

    for (int i = s; i < TT; i += 288) lse[i] = __logf(sumexp[n * TT + i]);

    bool skip = false;
    if (s < SS && s >= 2 && (s & 1)) {
        int zv  = labels[n * LL + (s >> 1)];
        int zm2 = labels[n * LL + ((s - 2) >> 1)];
        skip = (zv != 0) && (zv != zm2);
    }

    const float* gn = gbuf + (size_t)n * TT * SS;
    __syncthreads();
    if (s < SS) {
        alpha[0][s] = (s == 0) ? (gn[0] - lse[0])
                    : (s == 1) ? (gn[1] - lse[0]) : NEGV;
    }
    __syncthreads();

    int cur = 0;
    for (int t = 1; t < TT; ++t) {
        if (s < SS) {
            float a0 = alpha[cur][s];
            float nv;
            if (t < len) {
                float a1 = (s >= 1) ? alpha[cur][s - 1] : NEGV;
                float a2 = skip     ? alpha[cur][s - 2] : NEGV;
                float lp = gn[(size_t)t * SS + s] - lse[t];
                nv = lae(lae(a0, a1), a2) + lp;
            } else {
                nv = a0;
            }
            alpha[cur ^ 1][s] = nv;
        }
        __syncthreads();
        cur ^= 1;
    }

    if (s == 0) {
        int ib = 2 * lab_lens[n];
        nll[n] = -lae(alpha[cur][ib], alpha[cur][ib - 1]);
    }
}

// ---------------------------------------------------------------------------
// Kernel 4: mean NLL -> scalar output
// ---------------------------------------------------------------------------
__global__ void finalize(const float* __restrict__ nll, float* __restrict__ out) {
    if (threadIdx.x == 0) {
        float sum = 0.0f;
        for (int i = 0; i < NN; ++i) sum += nll[i];
        out[0] = sum / (float)NN;
    }
}

// ---------------------------------------------------------------------------
extern "C" void kernel_launch(void* const* d_in, const int* in_sizes, int n_in,
                              void* d_out, int out_size, void* d_ws, size_t ws_size,
                              hipStream_t stream) {
    const float* enc     = (const float*)d_in[0];  // (16,512,512)
    const float* W       = (const float*)d_in[1];  // (512,4096)
    const float* b       = (const float*)d_in[2];  // (4096)
    const int*   lens    = (const int*)d_in[3];    // (16)
    const int*   labels  = (const int*)d_in[4];    // (16,128)
    const int*   lablens = (const int*)d_in[5];    // (16)
    float* out = (float*)d_out;

    float* ws      = (float*)d_ws;
    float* sumexp  = ws;                 // 8192 floats
    float* nll     = ws + MM;            // 16 floats
    float* gbuf    = ws + MM + 16;       // 16*512*257 floats (~8.4 MB)

    zero_ws<<<(MM + 255) / 256, 256, 0, stream>>>(sumexp, MM);

    dim3 g1(VV / BN, MM / BM);           // 64 x 64 blocks
    gemm_sumexp<<<g1, 256, 0, stream>>>(enc, W, b, sumexp);

    dim3 g2(TT / GT, NN);                // 32 x 16 blocks
    gather_logits<<<g2, 256, 0, stream>>>(enc, W, b, labels, gbuf);

    ctc_forward<<<NN, 288, 0, stream>>>(gbuf, sumexp, labels, lens, lablens, nll);

    finalize<<<1, 32, 0, stream>>>(nll, out);
}